// OuterProductNetwork2_29145648070666
// MI455X (gfx1250) — compile-verified
//
#include <hip/hip_runtime.h>

#define NF      39
#define NPAIR   741          // 39*38/2
#define RPITCH  68           // dwords per padded LDS row: 64 data + 4 pad -> 64-bank conflict-free
#define THREADS 96           // 3 waves of 32 (wave32)

typedef float v2f __attribute__((ext_vector_type(2)));
typedef float v8f __attribute__((ext_vector_type(8)));
typedef int   v4i __attribute__((ext_vector_type(4)));

typedef __attribute__((address_space(1))) v4i g_v4i;   // global int4
typedef __attribute__((address_space(3))) v4i l_v4i;   // LDS int4

// Compile-time upper-triangular pair table: (row<<6)|col, row<39, col<39.
struct PairTab { unsigned short v[NPAIR]; };
static constexpr PairTab makePairs() {
  PairTab t{};
  int p = 0;
  for (int r = 0; r < NF; ++r)
    for (int c = r + 1; c < NF; ++c)
      t.v[p++] = (unsigned short)((r << 6) | c);
  return t;
}
__device__ constexpr PairTab kPairs = makePairs();

__global__ __launch_bounds__(THREADS)
void opn_fused_kernel(const float* __restrict__ x, float* __restrict__ out,
                      int nbatch) {
  __shared__ float tile[3 * 16 * RPITCH];  // 48 padded rows, 13,056 B
  __shared__ float sl[48];                 // per-field sums (39 valid)

  const int b    = blockIdx.x;
  const int tid  = threadIdx.x;
  const int lane = tid & 31;
  const int w    = tid >> 5;               // wave 0..2, handles rows w*16..w*16+15

  const float* xb = x + (size_t)b * (NF * 64);

  // ---- Stage 16 rows/wave into LDS: async global->LDS, b128 per lane ----
  // chunk q = i*32 + lane; row r = q/16, 16B-chunk j = q%16  => 512B coalesced / instr
  {
    const int r0 = lane >> 4;              // 0 or 1
    const int j  = lane & 15;
    int f = w * 16 + r0;                   // field (row) index
    const float* g = xb + f * 64 + j * 4;
    float*       l = &tile[(w * 16 + r0) * RPITCH + j * 4];
#pragma unroll
    for (int i = 0; i < 8; ++i) {
      if (f < NF) {
#if __has_builtin(__builtin_amdgcn_global_load_async_to_lds_b128)
        __builtin_amdgcn_global_load_async_to_lds_b128(
            (g_v4i*)g, (l_v4i*)l, /*offset=*/0, /*cpol=*/0);
#else
        *(float4*)l = *(const float4*)g;   // fallback: vmem load + ds store
#endif
      }
      g += 128;                            // +2 rows in global (contiguous)
      l += 2 * RPITCH;                     // +2 padded rows in LDS
      f += 2;
    }
  }
  __asm__ __volatile__("s_wait_asynccnt 0x0" ::: "memory");

  // ---- Reduce via WMMA: A = ones(16x4) => D[:,n] = colsum(B[:,n]) ----
  // B 4x16 f32 layout: lane l, vgpr v holds K = v + 2*(l/16), N = l%16.
  // Lane l therefore loads a float2 from LDS row (l%16) at dword 4c + 2*(l/16).
  v8f acc = {};
  const v2f aOnes = {1.0f, 1.0f};
  const float* lb = &tile[(w * 16 + (lane & 15)) * RPITCH + 2 * (lane >> 4)];
#pragma unroll
  for (int c = 0; c < 16; ++c) {           // K = 64 in 16 chained steps of 4
    v2f bv = *(const v2f*)(lb + 4 * c);    // ds_load_b64, bank-conflict-free
    acc = __builtin_amdgcn_wmma_f32_16x16x4_f32(
        false, aOnes, false, bv, (short)0, acc, false, false);
  }

  // D vgpr0, lanes 0..15 hold (M=0, N=lane) = colsum(lane)
  if (lane < 16) {
    sl[w * 16 + lane] = acc[0];            // rows >= NF write garbage slots, never read
  }
  __syncthreads();

  // ---- 741 pairwise products, coalesced store ----
  float* ob = out + (size_t)b * NPAIR;
  for (int p = tid; p < NPAIR; p += THREADS) {
    const unsigned pr = kPairs.v[p];
    ob[p] = sl[pr >> 6] * sl[pr & 63];
  }
}

extern "C" void kernel_launch(void* const* d_in, const int* in_sizes, int n_in,
                              void* d_out, int out_size, void* d_ws, size_t ws_size,
                              hipStream_t stream) {
  const float* x = (const float*)d_in[0];
  float* out = (float*)d_out;
  const int nbatch = in_sizes[0] / (NF * 64);  // 16384 for the reference shapes
  opn_fused_kernel<<<nbatch, THREADS, 0, stream>>>(x, out, nbatch);
}